// ConvGraph_31284541784246
// MI455X (gfx1250) — compile-verified
//
#include <hip/hip_runtime.h>
#include <hip/hip_bf16.h>

#define N_NODES 20000
#define HIDDEN  128

typedef float v2f __attribute__((ext_vector_type(2)));
typedef float v8f __attribute__((ext_vector_type(8)));

// ---------------------------------------------------------------------------
// Kernel 1: zero the scratch (summed accumulator + degree counts)
// ---------------------------------------------------------------------------
__global__ void sage_zero_kernel(float* __restrict__ p, int n) {
    int i = blockIdx.x * blockDim.x + threadIdx.x;
    if (i < n) p[i] = 0.0f;
}

// ---------------------------------------------------------------------------
// Kernel 2: edge scatter-add. One wave32 per edge. Lane l handles feature
// columns l, l+32, l+64, l+96 -> 4 coalesced global_atomic_add_f32 per lane.
// All traffic is L2-resident (features 10.24 MB, summed 10.24 MB << 192 MB L2).
// ---------------------------------------------------------------------------
__global__ __launch_bounds__(256) void sage_scatter_kernel(
    const long long* __restrict__ edge_index,   // [2, E] int64, row-major
    const float* __restrict__ feat,             // [N, 128]
    float* __restrict__ summed,                 // [N, 128] accumulator
    float* __restrict__ cnt,                    // [N] degree counts
    int E)
{
    int wave = (blockIdx.x * blockDim.x + threadIdx.x) >> 5;
    int lane = threadIdx.x & 31;
    if (wave >= E) return;

    long long s = edge_index[wave];      // src
    long long d = edge_index[E + wave];  // dst
    if ((unsigned long long)s >= (unsigned long long)N_NODES ||
        (unsigned long long)d >= (unsigned long long)N_NODES) return;

    const float* fs = feat   + (size_t)s * HIDDEN;
    float*       sd = summed + (size_t)d * HIDDEN;
#pragma unroll
    for (int i = 0; i < 4; ++i)
        atomicAdd(sd + lane + 32 * i, fs[lane + 32 * i]);
    if (lane == 0) atomicAdd(cnt + d, 1.0f);
}

// ---------------------------------------------------------------------------
// Kernel 3: out = (summed/max(cnt,1)) @ Wl^T + b + x @ Wr^T using fp32 WMMA.
// Block = 256 threads = 8 waves = one 16-row M-tile; wave w owns N-tile w.
// Per ISA 7.12.2 (32-bit operands, wave32): for A(16x4) and B(4x16),
// lane group g = lane>>4, row/col index r = lane&15, VGPR v -> K = 2g + v.
// C/D(16x16): VGPR v, group g -> M = v + 8g, N = r.
// ---------------------------------------------------------------------------
__global__ __launch_bounds__(256) void sage_gemm_kernel(
    const float* __restrict__ summed,  // [N,128]
    const float* __restrict__ cnt,     // [N]
    const float* __restrict__ x,       // [N,128]
    const float* __restrict__ Wl,      // [128,128] row-major
    const float* __restrict__ bl,      // [128]
    const float* __restrict__ Wr,      // [128,128]
    float* __restrict__ out)           // [N,128]
{
    const int lane = threadIdx.x & 31;
    const int wave = threadIdx.x >> 5;      // 0..7 -> N tile
    const int g    = lane >> 4;             // lane group (0/1)
    const int r    = lane & 15;             // row-in-tile / col-in-tile
    const int row  = blockIdx.x * 16 + r;   // A-matrix row this lane feeds
    const int ncol = wave * 16 + r;         // B column == weight-matrix row

    // fold the mean into the A load: scale summed by 1/max(cnt,1)
    const float invc = 1.0f / fmaxf(cnt[row], 1.0f);

    const float* pa1 = summed + (size_t)row  * HIDDEN + 2 * g;
    const float* pa2 = x      + (size_t)row  * HIDDEN + 2 * g;
    const float* pb1 = Wl     + (size_t)ncol * HIDDEN + 2 * g;  // B[k][n]=Wl[n][k]
    const float* pb2 = Wr     + (size_t)ncol * HIDDEN + 2 * g;

    v8f acc1 = {0.f, 0.f, 0.f, 0.f, 0.f, 0.f, 0.f, 0.f};
    v8f acc2 = {0.f, 0.f, 0.f, 0.f, 0.f, 0.f, 0.f, 0.f};

#pragma unroll
    for (int k = 0; k < HIDDEN; k += 4) {
        v2f a1 = *(const v2f*)(pa1 + k);
        a1.x *= invc;
        a1.y *= invc;
        v2f b1 = *(const v2f*)(pb1 + k);
        acc1 = __builtin_amdgcn_wmma_f32_16x16x4_f32(
            false, a1, false, b1, (short)0, acc1, false, false);

        v2f a2 = *(const v2f*)(pa2 + k);
        v2f b2 = *(const v2f*)(pb2 + k);
        acc2 = __builtin_amdgcn_wmma_f32_16x16x4_f32(
            false, a2, false, b2, (short)0, acc2, false, false);
    }

    const float bias = bl[ncol];
#pragma unroll
    for (int v = 0; v < 8; ++v) {
        int orow = blockIdx.x * 16 + v + 8 * g;   // D: M = v + 8g
        out[(size_t)orow * HIDDEN + ncol] = acc1[v] + acc2[v] + bias;
    }
}

// ---------------------------------------------------------------------------
// Launch: zero scratch -> edge scatter -> fused mean + dual WMMA GEMM.
// d_ws usage: [0, N*128) summed accumulator, [N*128, N*128+N) degree counts.
// ---------------------------------------------------------------------------
extern "C" void kernel_launch(void* const* d_in, const int* in_sizes, int n_in,
                              void* d_out, int out_size, void* d_ws, size_t ws_size,
                              hipStream_t stream) {
    const float*     feat = (const float*)d_in[0];
    const long long* ei   = (const long long*)d_in[1];   // int64 per reference
    const float*     Wl   = (const float*)d_in[2];
    const float*     bl   = (const float*)d_in[3];
    const float*     Wr   = (const float*)d_in[4];
    float*           out  = (float*)d_out;

    float* summed = (float*)d_ws;
    float* cnt    = summed + (size_t)N_NODES * HIDDEN;

    const int E = in_sizes[1] / 2;   // 640000

    // 1) zero summed + cnt (deterministic per call; harness does not re-poison)
    int ztotal = N_NODES * HIDDEN + N_NODES;
    sage_zero_kernel<<<(ztotal + 255) / 256, 256, 0, stream>>>(summed, ztotal);

    // 2) scatter-add: one wave32 per edge
    long long sthreads = (long long)E * 32;
    sage_scatter_kernel<<<(sthreads + 255) / 256, 256, 0, stream>>>(
        ei, feat, summed, cnt, E);

    // 3) fused mean + (agg @ Wl^T + b + x @ Wr^T), fp32 WMMA, 1250 M-tiles
    sage_gemm_kernel<<<N_NODES / 16, 256, 0, stream>>>(
        summed, cnt, feat, Wl, bl, Wr, out);
}